// Determinant_56556129354389
// MI455X (gfx1250) — compile-verified
//
#include <hip/hip_runtime.h>

#define NS     96
#define STRIDE 97          // odd stride -> conflict-free column access on 64 banks
#define NBLK   6           // 96 / 16
#define TPB    256         // 8 wave32

typedef __attribute__((ext_vector_type(2))) float v2f;
typedef __attribute__((ext_vector_type(8))) float v8f;

__global__ __launch_bounds__(TPB)
void det96_lu_wmma(const int* __restrict__ x,
                   const float* __restrict__ U,
                   float* __restrict__ out)
{
    __shared__ float As[NS * STRIDE];   // 37,248 B
    __shared__ int   idx[NS];
    __shared__ float redV[TPB / 32];
    __shared__ int   redI[TPB / 32];
    __shared__ int   pivRow;

    const int b    = blockIdx.x;
    const int tid  = threadIdx.x;
    const int lane = tid & 31;
    const int wave = tid >> 5;

    double det = 1.0;   // only thread 0's copy is meaningful

    // ---- 1. fermion index list: {i : x_i>0} then {96+i : x_i==0} ----
    if (tid == 0) {
        const int* xb = x + (long)b * NS;
        int cnt = 0;
        for (int i = 0; i < NS; ++i) if (xb[i] >  0) idx[cnt++] = i;
        for (int i = 0; i < NS; ++i) if (xb[i] <= 0) idx[cnt++] = NS + i;
    }
    __syncthreads();

    // ---- 2. gather rows of U into LDS ----
    for (int e = tid; e < NS * NS; e += TPB) {
        int r = e / NS;
        int c = e - r * NS;
        As[r * STRIDE + c] = U[idx[r] * NS + c];
    }
    __syncthreads();

    // ---- 3. blocked LU with partial pivoting ----
    for (int kb = 0; kb < NBLK; ++kb) {
        const int K0 = kb * 16;
        const int KE = K0 + 16;

        // -------- panel factorization, columns K0..KE-1 --------
        for (int j = K0; j < KE; ++j) {
            // argmax |As[i][j]| over i = j..NS-1
            float v = -1.0f; int vi = j;
            for (int i = j + tid; i < NS; i += TPB) {
                float a = fabsf(As[i * STRIDE + j]);
                if (a > v) { v = a; vi = i; }
            }
            for (int off = 16; off > 0; off >>= 1) {
                float ov = __shfl_down(v,  off);
                int   oi = __shfl_down(vi, off);
                if (ov > v) { v = ov; vi = oi; }
            }
            if (lane == 0) { redV[wave] = v; redI[wave] = vi; }
            __syncthreads();
            if (tid == 0) {
                float bv = redV[0]; int bi = redI[0];
                for (int w = 1; w < TPB / 32; ++w)
                    if (redV[w] > bv) { bv = redV[w]; bi = redI[w]; }
                pivRow = bi;
            }
            __syncthreads();
            const int p = pivRow;

            // full-row swap (covers L part, panel and trailing columns)
            if (p != j) {
                for (int c = tid; c < NS; c += TPB) {
                    float t0 = As[j * STRIDE + c];
                    As[j * STRIDE + c] = As[p * STRIDE + c];
                    As[p * STRIDE + c] = t0;
                }
            }
            __syncthreads();

            const float piv = As[j * STRIDE + j];
            if (tid == 0) det *= (p != j) ? -(double)piv : (double)piv;
            const float rp = (piv != 0.0f) ? (1.0f / piv) : 0.0f;

            // negated multipliers + elimination over panel columns (1 thread/row)
            for (int i = j + 1 + tid; i < NS; i += TPB) {
                float m = -As[i * STRIDE + j] * rp;
                As[i * STRIDE + j] = m;
                for (int c = j + 1; c < KE; ++c)
                    As[i * STRIDE + c] = fmaf(m, As[j * STRIDE + c], As[i * STRIDE + c]);
            }
            __syncthreads();

            // diagonal-block rows: forward-substitute trailing columns
            // (produces U12 = L11^{-1} A12 incrementally)
            const int nr = KE - (j + 1);
            const int nc = NS - KE;
            for (int e2 = tid; e2 < nr * nc; e2 += TPB) {
                int i = j + 1 + e2 / nc;
                int c = KE + (e2 - (e2 / nc) * nc);
                As[i * STRIDE + c] =
                    fmaf(As[i * STRIDE + j], As[j * STRIDE + c], As[i * STRIDE + c]);
            }
            __syncthreads();
        }

        // -------- Schur update A22 += M21 * U12 via V_WMMA_F32_16X16X4_F32 --------
        const int nt    = (NS / 16) - (kb + 1);
        const int total = nt * nt;
        for (int t = wave; t < total; t += TPB / 32) {   // wave-uniform: EXEC stays all-ones
            const int rt = kb + 1 + t / nt;
            const int ct = kb + 1 + (t - (t / nt) * nt);
            const int R0 = rt * 16, C0 = ct * 16;
            const int m  = lane & 15;
            const int h  = lane >> 4;                    // half-wave selector

            // C/D layout: VGPR v, lane L -> M = v + 8*(L/16), N = L%16
            v8f cacc;
            #pragma unroll
            for (int v = 0; v < 8; ++v)
                cacc[v] = As[(R0 + v + 8 * h) * STRIDE + C0 + m];

            #pragma unroll
            for (int kk = 0; kk < 4; ++kk) {
                const int kcol = K0 + kk * 4 + 2 * h;
                // A 16x4 layout: lane M = L%16; VGPR v -> K = 2*(L/16) + v
                v2f afr, bfr;
                afr[0] = As[(R0 + m) * STRIDE + kcol + 0];
                afr[1] = As[(R0 + m) * STRIDE + kcol + 1];
                // B 4x16 layout: VGPR v -> K = 2*(L/16) + v, N = L%16
                bfr[0] = As[(kcol + 0) * STRIDE + C0 + m];
                bfr[1] = As[(kcol + 1) * STRIDE + C0 + m];
                cacc = __builtin_amdgcn_wmma_f32_16x16x4_f32(
                           false, afr, false, bfr, (short)0, cacc, false, false);
            }

            #pragma unroll
            for (int v = 0; v < 8; ++v)
                As[(R0 + v + 8 * h) * STRIDE + C0 + m] = cacc[v];
        }
        __syncthreads();
    }

    if (tid == 0) out[b] = (float)det;
}

extern "C" void kernel_launch(void* const* d_in, const int* in_sizes, int n_in,
                              void* d_out, int out_size, void* d_ws, size_t ws_size,
                              hipStream_t stream)
{
    (void)n_in; (void)out_size; (void)d_ws; (void)ws_size;
    const int*   x = (const int*)d_in[0];     // (BATCH, 96) int32
    const float* U = (const float*)d_in[1];   // (192, 96) fp32
    float* out = (float*)d_out;               // (BATCH,) fp32
    const int batch = in_sizes[0] / NS;       // 4096
    det96_lu_wmma<<<batch, TPB, 0, stream>>>(x, U, out);
}